// GAT_39170101739643
// MI455X (gfx1250) — compile-verified
//
#include <hip/hip_runtime.h>
#include <hip/hip_bf16.h>
#include <math.h>

// ---------------------------------------------------------------------------
// GAT forward (2x GATConv, edge_dim, self-loops) for MI455X / gfx1250.
// fp32 WMMA (V_WMMA_F32_16X16X4_F32) for both node-projection GEMMs with the
// B panel staged into LDS by the Tensor Data Mover (tensor_load_to_lds +
// s_wait_tensorcnt); segment-softmax via float-ordered atomicMax + atomicAdd.
// ---------------------------------------------------------------------------

typedef __attribute__((ext_vector_type(2))) float v2f;
typedef __attribute__((ext_vector_type(8))) float v8f;

#if __has_builtin(__builtin_amdgcn_tensor_load_to_lds)
#define HAVE_TDM 1
typedef __attribute__((ext_vector_type(4))) unsigned int tdm_v4u;
typedef __attribute__((ext_vector_type(8))) int          tdm_v8i;
typedef __attribute__((ext_vector_type(4))) int          tdm_v4i;
#endif

#define NEG_SLOPE 0.2f
#define SM_EPS    1e-16f

static __device__ __forceinline__ int ord_map(float f) {
    int i = __float_as_int(f);
    return i >= 0 ? i : (i ^ 0x7FFFFFFF);
}
static __device__ __forceinline__ float ord_unmap(int i) {
    return __int_as_float(i >= 0 ? i : (i ^ 0x7FFFFFFF));
}

// ------------------------- fill helpers ------------------------------------
__global__ void k_fill_f32(float* __restrict__ p, float v, int n) {
    int t = blockIdx.x * blockDim.x + threadIdx.x;
    if (t < n) p[t] = v;
}
__global__ void k_fill_i32(int* __restrict__ p, int v, int n) {
    int t = blockIdx.x * blockDim.x + threadIdx.x;
    if (t < n) p[t] = v;
}

// ------------------- WMMA GEMM: C[M,LDB] = A[M,128] * B[128,LDB] ----------
// One wave per 16x16 output tile; 8 waves / block.  The whole B panel
// (128 x LDB fp32) is DMA'd into LDS once per block via the TDM, then each
// wave runs K=128 as 32 x V_WMMA_F32_16X16X4_F32 with immediate-offset
// global_load_b64 (A) and ds_load_b32 (B) operand feeds.
// A layout (16x4 f32): lanes 0-15 K=k0+{0,1}; lanes 16-31 K=k0+{2,3}.
// B layout (4x16 f32): VGPR0 = K {k0, k0+2}, VGPR1 = {k0+1, k0+3}.
// C/D layout: VGPR r -> (M = r + 8*half, N = l).
template <int LDB>
__global__ void __launch_bounds__(256)
k_wmma_gemm_k128(const float* __restrict__ A, const float* __restrict__ B,
                 float* __restrict__ C, int M, int ntn) {
    __shared__ float Bs[128 * LDB];

    const int wave = threadIdx.x >> 5;
    const int lane = threadIdx.x & 31;

    // ---- stage B[128, LDB] into LDS ----
#if defined(HAVE_TDM)
    if (threadIdx.x < 32) {                       // wave 0 issues the DMA
        const unsigned long long ga = (unsigned long long)(const void*)B;
        const unsigned lds0 = (unsigned)(unsigned long long)(void*)&Bs[0];
        tdm_v4u g0;
        g0[0] = 1u;                               // count=1, user descriptor
        g0[1] = lds0;                             // lds_addr
        g0[2] = (unsigned)ga;                     // global_addr[31:0]
        g0[3] = (unsigned)((ga >> 32) & 0x01FFFFFFu) | (2u << 30); // addr[56:32] | type=2
        tdm_v8i g1;
        g1[0] = (int)(2u << 16);                  // wg_mask=0, data_size=4B
        g1[1] = (int)(((unsigned)LDB & 0xFFFFu) << 16);   // tensor_dim0[15:0]
        g1[2] = (int)(128u << 16);                // tensor_dim0 hi | tensor_dim1=128
        g1[3] = (int)((unsigned)LDB << 16);       // tensor_dim1 hi | tile_dim0=LDB
        g1[4] = 128;                              // tile_dim1=128, tile_dim2=0
        g1[5] = LDB;                              // tensor_dim0_stride lo
        g1[6] = 0;
        g1[7] = 0;
        tdm_v4i gz = {0, 0, 0, 0};
#if __clang_major__ >= 23
        tdm_v8i gz8 = {0, 0, 0, 0, 0, 0, 0, 0};
        __builtin_amdgcn_tensor_load_to_lds(g0, g1, gz, gz, gz8, 0);
#else
        __builtin_amdgcn_tensor_load_to_lds(g0, g1, gz, gz, 0);
#endif
        __builtin_amdgcn_s_wait_tensorcnt((short)0);
    }
#else
    for (int i = threadIdx.x; i < 128 * LDB; i += 256) Bs[i] = B[i];
#endif
    __syncthreads();

    const int tile = blockIdx.x * 8 + wave;
    const int ntm  = M >> 4;                      // M is a multiple of 16
    if (tile >= ntm * ntn) return;                // wave-uniform: EXEC stays full
    const int tm = tile / ntn;
    const int tn = tile - tm * ntn;
    const int half = lane >> 4;
    const int l    = lane & 15;

    const float* __restrict__ Ap = A + (size_t)(tm * 16 + l) * 128 + 2 * half;
    const float* __restrict__ Bp = &Bs[2 * half * LDB + tn * 16 + l];

    v8f acc = {};
#pragma unroll
    for (int k0 = 0; k0 < 128; k0 += 4) {
        const float2 a2 = *(const float2*)(Ap + k0);        // imm-offset b64 load
        v2f a; a.x = a2.x; a.y = a2.y;
        v2f b; b.x = Bp[k0 * LDB];                          // imm-offset ds loads
               b.y = Bp[(k0 + 1) * LDB];
        acc = __builtin_amdgcn_wmma_f32_16x16x4_f32(
                  /*neg_a=*/false, a, /*neg_b=*/false, b,
                  /*c_mod=*/(short)0, acc, /*reuse_a=*/false, /*reuse_b=*/false);
    }
#pragma unroll
    for (int r = 0; r < 8; ++r) {
        const int m = tm * 16 + r + 8 * half;
        C[(size_t)m * LDB + tn * 16 + l] = acc[r];
    }
}

// --------- per-node attention dots: a_src/a_dst[n,h] = <h_row, att> --------
__global__ void k_att_pre(const float* __restrict__ hp,
                          const float* __restrict__ av_s,
                          const float* __restrict__ av_d,
                          float* __restrict__ osrc, float* __restrict__ odst,
                          int N, int H, int C, int ld) {
    int t = blockIdx.x * blockDim.x + threadIdx.x;
    if (t >= N * H) return;
    int n = t / H, h = t - n * H;
    const float* row = hp + (size_t)n * ld + h * C;
    float ss = 0.f, sd = 0.f;
    for (int c = 0; c < C; ++c) {
        float v = row[c];
        ss = fmaf(v, av_s[h * C + c], ss);
        sd = fmaf(v, av_d[h * C + c], sd);
    }
    osrc[t] = ss;
    odst[t] = sd;
}

// ---- we_att[d,h] = sum_c We[d, h*C+c] * ae[h*C+c]  (collapses edge proj) ---
__global__ void k_weatt(const float* __restrict__ We, const float* __restrict__ ae,
                        float* __restrict__ out, int H, int C) {
    int t = threadIdx.x;
    if (t >= 3 * H) return;
    int d = t / H, h = t - d * H;
    float s = 0.f;
    for (int c = 0; c < C; ++c) s = fmaf(We[(size_t)d * H * C + h * C + c], ae[h * C + c], s);
    out[d * H + h] = s;
}

// --------- pass A: alpha = leakyrelu(asrc[s]+adst[d]+ea.weatt); seg-max ----
__global__ void k_alpha_max(const int* __restrict__ ei, const float* __restrict__ ef,
                            const float* __restrict__ asrc, const float* __restrict__ adst,
                            const float* __restrict__ weatt,
                            float* __restrict__ alpha, int* __restrict__ amax,
                            int E, int N, int H) {
    int t = blockIdx.x * blockDim.x + threadIdx.x;
    int tot = (E + N) * H;
    if (t >= tot) return;
    int e = t / H, h = t - e * H;
    int s, d;
    float ed = 0.f;
    if (e < E) {
        s = ei[e];
        d = ei[E + e];
        const float* ea = ef + (size_t)e * 3;
        for (int k = 0; k < 3; ++k) ed = fmaf(ea[k], weatt[k * H + h], ed);
    } else {                     // self-loop: src=dst, edge_attr = 0
        s = d = e - E;
    }
    float a = asrc[s * H + h] + adst[d * H + h] + ed;
    a = a > 0.f ? a : NEG_SLOPE * a;
    alpha[t] = a;
    atomicMax(&amax[d * H + h], ord_map(a));
}

// --------- pass B: alpha = exp(alpha - amax[dst]); denom += alpha ----------
__global__ void k_exp_sum(const int* __restrict__ ei,
                          float* __restrict__ alpha,
                          const int* __restrict__ amax, float* __restrict__ denom,
                          int E, int N, int H) {
    int t = blockIdx.x * blockDim.x + threadIdx.x;
    int tot = (E + N) * H;
    if (t >= tot) return;
    int e = t / H, h = t - e * H;
    int d = (e < E) ? ei[E + e] : (e - E);
    float mx = ord_unmap(amax[d * H + h]);
    float ex = expf(alpha[t] - mx);
    alpha[t] = ex;
    atomicAdd(&denom[d * H + h], ex);
}

// --------- pass C layer1: out[dst] += h1[src] * alpha/denom  (H=2,C=64) ----
// one thread = one (edge, head, 4-channel chunk); float4 gather + 4 atomics.
__global__ void k_scatter1(const int* __restrict__ ei, const float* __restrict__ h1,
                           const float* __restrict__ alpha, const float* __restrict__ denom,
                           float* __restrict__ out, int E, int N) {
    int t = blockIdx.x * blockDim.x + threadIdx.x;
    int tot = (E + N) * 32;                 // 2 heads * 16 chunks
    if (t >= tot) return;
    int chunk = t & 15;
    int h     = (t >> 4) & 1;
    int e     = t >> 5;
    int s, d;
    if (e < E) { s = ei[e]; d = ei[E + e]; } else { s = d = e - E; }
    float w = alpha[e * 2 + h] / (denom[d * 2 + h] + SM_EPS);
    const float4 hv = *(const float4*)(h1 + (size_t)s * 128 + h * 64 + chunk * 4);
    float* o = out + (size_t)d * 128 + h * 64 + chunk * 4;
    atomicAdd(o + 0, hv.x * w);
    atomicAdd(o + 1, hv.y * w);
    atomicAdd(o + 2, hv.z * w);
    atomicAdd(o + 3, hv.w * w);
}

// --------- pass C layer2: out[dst,0:2] += h2p[src,0:2] * alpha/denom -------
__global__ void k_scatter2(const int* __restrict__ ei, const float* __restrict__ h2p,
                           const float* __restrict__ alpha, const float* __restrict__ denom,
                           float* __restrict__ out, int E, int N) {
    int e = blockIdx.x * blockDim.x + threadIdx.x;
    if (e >= E + N) return;
    int s, d;
    if (e < E) { s = ei[e]; d = ei[E + e]; } else { s = d = e - E; }
    float w = alpha[e] / (denom[d] + SM_EPS);
    const float2 hv = *(const float2*)(h2p + (size_t)s * 16);
    atomicAdd(&out[(size_t)d * 2 + 0], hv.x * w);
    atomicAdd(&out[(size_t)d * 2 + 1], hv.y * w);
}

// --------- finalize layer1: h2 = elu(out1 + b1) in place -------------------
__global__ void k_elu_bias(float* __restrict__ o, const float* __restrict__ b, int total) {
    int t = blockIdx.x * blockDim.x + threadIdx.x;
    if (t >= total) return;
    float v = o[t] + b[t & 127];
    o[t] = v > 0.f ? v : (expf(v) - 1.f);
}

// --------- finalize layer2: out += b2 --------------------------------------
__global__ void k_bias2(float* __restrict__ o, const float* __restrict__ b, int total) {
    int t = blockIdx.x * blockDim.x + threadIdx.x;
    if (t >= total) return;
    o[t] += b[t & 1];
}

// --------- pad W2 [128,2] -> W2p [128,16] (zeros) for the WMMA GEMM --------
__global__ void k_padw2(const float* __restrict__ W2, float* __restrict__ W2p) {
    int t = blockIdx.x * blockDim.x + threadIdx.x;   // 128*16
    if (t >= 128 * 16) return;
    int r = t >> 4, c = t & 15;
    W2p[t] = (c < 2) ? W2[r * 2 + c] : 0.f;
}

// ---------------------------------------------------------------------------
extern "C" void kernel_launch(void* const* d_in, const int* in_sizes, int n_in,
                              void* d_out, int out_size, void* d_ws, size_t ws_size,
                              hipStream_t stream) {
    const float* x   = (const float*)d_in[0];
    const int*   ei  = (const int*)  d_in[1];
    const float* ef  = (const float*)d_in[2];
    const float* W1  = (const float*)d_in[3];
    const float* We1 = (const float*)d_in[4];
    const float* as1 = (const float*)d_in[5];
    const float* ad1 = (const float*)d_in[6];
    const float* ae1 = (const float*)d_in[7];
    const float* b1  = (const float*)d_in[8];
    const float* W2  = (const float*)d_in[9];
    const float* We2 = (const float*)d_in[10];
    const float* as2 = (const float*)d_in[11];
    const float* ad2 = (const float*)d_in[12];
    const float* ae2 = (const float*)d_in[13];
    const float* b2  = (const float*)d_in[14];
    float* out = (float*)d_out;

    const int N  = in_sizes[0] / 128;   // 100000
    const int E  = in_sizes[1] / 2;     // 1600000
    const int ET = E + N;               // with self loops

    // ---- workspace carving (256B aligned slabs) ----
    char* wp = (char*)d_ws;
    auto carve = [&](size_t bytes) -> void* {
        void* p = (void*)wp;
        wp += (bytes + 255) & ~(size_t)255;
        return p;
    };
    float* h1     = (float*)carve((size_t)N * 128 * 4);  // x @ W1
    float* out1   = (float*)carve((size_t)N * 128 * 4);  // layer1 aggregate -> h2
    float* h2p    = (float*)carve((size_t)N * 16 * 4);   // h2 @ W2p (padded to 16 cols)
    float* a1s    = (float*)carve((size_t)N * 2 * 4);
    float* a1d    = (float*)carve((size_t)N * 2 * 4);
    float* a2s    = (float*)carve((size_t)N * 4);
    float* a2d    = (float*)carve((size_t)N * 4);
    int*   amax1  = (int*)  carve((size_t)N * 2 * 4);
    float* den1   = (float*)carve((size_t)N * 2 * 4);
    int*   amax2  = (int*)  carve((size_t)N * 4);
    float* den2   = (float*)carve((size_t)N * 4);
    float* alphab = (float*)carve((size_t)ET * 2 * 4);   // reused by layer2 (needs ET)
    float* weatt1 = (float*)carve(6 * 4);
    float* weatt2 = (float*)carve(3 * 4);
    float* W2p    = (float*)carve(128 * 16 * 4);

    const int TB = 256;
    auto blk = [](long long n, int t) { return (unsigned)((n + t - 1) / t); };

    // ---- init ----
    k_fill_i32<<<blk(N * 2, TB), TB, 0, stream>>>(amax1, (int)0x80000000, N * 2);
    k_fill_f32<<<blk(N * 2, TB), TB, 0, stream>>>(den1, 0.f, N * 2);
    k_fill_i32<<<blk(N, TB), TB, 0, stream>>>(amax2, (int)0x80000000, N);
    k_fill_f32<<<blk(N, TB), TB, 0, stream>>>(den2, 0.f, N);
    k_fill_f32<<<blk((long long)N * 128, TB), TB, 0, stream>>>(out1, 0.f, N * 128);
    k_fill_f32<<<blk(N * 2, TB), TB, 0, stream>>>(out, 0.f, N * 2);

    // ---- collapse edge projections against att_edge ----
    k_weatt<<<1, 32, 0, stream>>>(We1, ae1, weatt1, 2, 64);
    k_weatt<<<1, 32, 0, stream>>>(We2, ae2, weatt2, 1, 2);

    // ================= layer 1 =================
    {   // h1 = x @ W1  (WMMA + TDM-staged B panel, ntn=8)
        int tiles = (N >> 4) * 8;
        k_wmma_gemm_k128<128><<<blk(tiles, 8), TB, 0, stream>>>(x, W1, h1, N, 8);
    }
    k_att_pre<<<blk(N * 2, TB), TB, 0, stream>>>(h1, as1, ad1, a1s, a1d, N, 2, 64, 128);
    k_alpha_max<<<blk((long long)ET * 2, TB), TB, 0, stream>>>(ei, ef, a1s, a1d, weatt1,
                                                               alphab, amax1, E, N, 2);
    k_exp_sum<<<blk((long long)ET * 2, TB), TB, 0, stream>>>(ei, alphab, amax1, den1, E, N, 2);
    k_scatter1<<<blk((long long)ET * 32, TB), TB, 0, stream>>>(ei, h1, alphab, den1, out1, E, N);
    k_elu_bias<<<blk((long long)N * 128, TB), TB, 0, stream>>>(out1, b1, N * 128);  // out1 -> h2

    // ================= layer 2 =================
    k_padw2<<<blk(128 * 16, TB), TB, 0, stream>>>(W2, W2p);
    {   // h2p = h2 @ W2p (WMMA + TDM-staged B panel, single 16-wide column tile)
        int tiles = (N >> 4);
        k_wmma_gemm_k128<16><<<blk(tiles, 8), TB, 0, stream>>>(out1, W2p, h2p, N, 1);
    }
    k_att_pre<<<blk(N, TB), TB, 0, stream>>>(h2p, as2, ad2, a2s, a2d, N, 1, 2, 16);
    k_alpha_max<<<blk(ET, TB), TB, 0, stream>>>(ei, ef, a2s, a2d, weatt2, alphab, amax2, E, N, 1);
    k_exp_sum<<<blk(ET, TB), TB, 0, stream>>>(ei, alphab, amax2, den2, E, N, 1);
    k_scatter2<<<blk(ET, TB), TB, 0, stream>>>(ei, h2p, alphab, den2, out, E, N);
    k_bias2<<<blk(N * 2, TB), TB, 0, stream>>>(out, b2, N * 2);

    (void)n_in; (void)out_size; (void)ws_size;
}